// SparseResBlock_45844480917741
// MI455X (gfx1250) — compile-verified
//
#include <hip/hip_runtime.h>

// SparseResBlock on gfx1250: gather-GEMM with V_WMMA_F32_16X16X4_F32,
// double-buffered LDS staging, pre-paired B layout for b64 fragment loads.
// conv: out[n] = sum_{k<27} x[nbr[n,k]] @ W[k]  (W[k] is 64x64, fp32)
// Stage 1: h1 = relu(bn1(conv(x, W1)))            -> d_ws
// Stage 2: out = relu(bn2(conv(h1, W2)) + x)      -> d_out

typedef __attribute__((ext_vector_type(2))) float v2f;
typedef __attribute__((ext_vector_type(4))) float v4f;
typedef __attribute__((ext_vector_type(8))) float v8f;

#define NNODES  200000
#define CCH     64     // channels (Cin == Cout)
#define KVOL    27     // 3x3x3 kernel volume
#define MT      64     // nodes per workgroup tile
#define LDA     68     // A LDS row stride (floats): banks (4r+c)%64, conflict-free frags
#define SBST    80     // B LDS row-pair stride (float2): 2*80%64==32 -> disjoint bank halves
#define THREADS 512    // 16 wave32s: 4 M-subtiles x 4 N-subtiles of 16x16

__global__ __launch_bounds__(THREADS)
void sparse_conv_bn_kernel(const float* __restrict__ src,   // gather source [N, 64]
                           const int*   __restrict__ nbr,   // [N, 27]
                           const float* __restrict__ W,     // [27, 64, 64] (Cin x Cout)
                           const float* __restrict__ g,
                           const float* __restrict__ b,
                           const float* __restrict__ m,
                           const float* __restrict__ v,
                           const float* __restrict__ resid, // residual source (may be null)
                           float*       __restrict__ dst,   // [N, 64]
                           int resid_flag)
{
    __shared__ float sa[2][MT * LDA];        // gathered activation tiles (double buffered)
    __shared__ v2f   sb[2][32 * SBST];       // W[k] tiles, K-row pairs interleaved per column

    const int tid  = threadIdx.x;
    const int base = blockIdx.x * MT;

    const int lane  = tid & 31;
    const int wv    = tid >> 5;           // wave 0..15
    const int mi    = (wv >> 2) * 16;     // M subtile origin
    const int ni    = (wv & 3) * 16;      // N subtile origin (channel block)
    const int lhalf = lane >> 4;          // 0: lanes 0-15, 1: lanes 16-31
    const int lm    = lane & 15;

    // A staging: thread owns 8 floats of row ra
    const int ra = tid >> 3;              // 0..63
    const int ca = (tid & 7) * 8;         // 0,8,...,56
    // B staging: thread owns 4 columns of K-row-pair rp
    const int rp = tid >> 4;              // 0..31
    const int cb = (tid & 15) * 4;        // 0,4,...,60

    // BN coefficients for this lane's output channel (loads overlap main loop)
    const int   c  = ni + lm;
    const float sc = g[c] * rsqrtf(v[c] + 1e-5f);
    const float tc = b[c] - m[c] * sc;

    // ---- prologue: prefetch k=0 tiles into registers ----
    v4f a0, a1, w0, w1;
    {
        const int idx = nbr[(base + ra) * KVOL];
        const float* srow = src + (size_t)idx * CCH + ca;
        a0 = *(const v4f*)(srow);
        a1 = *(const v4f*)(srow + 4);
        const float* wr = W + (size_t)(2 * rp) * CCH + cb;
        w0 = *(const v4f*)(wr);
        w1 = *(const v4f*)(wr + CCH);
    }

    v8f acc = {0.f, 0.f, 0.f, 0.f, 0.f, 0.f, 0.f, 0.f};

    for (int k = 0; k < KVOL; ++k) {
        const int buf = k & 1;

        // ---- commit prefetched tiles to LDS ----
        *(v4f*)&sa[buf][ra * LDA + ca]     = a0;
        *(v4f*)&sa[buf][ra * LDA + ca + 4] = a1;
        // interleave K-row pair per column: sb[rp][c] = {W[2rp][c], W[2rp+1][c]}
        const v4f plo = {w0.x, w1.x, w0.y, w1.y};
        const v4f phi = {w0.z, w1.z, w0.w, w1.w};
        *(v4f*)&sb[buf][rp * SBST + cb]     = plo;
        *(v4f*)&sb[buf][rp * SBST + cb + 2] = phi;

        // ---- prefetch next k (hides gather latency behind WMMA chain) ----
        if (k + 1 < KVOL) {
            const int idx = nbr[(base + ra) * KVOL + (k + 1)];
            const float* srow = src + (size_t)idx * CCH + ca;
            a0 = *(const v4f*)(srow);
            a1 = *(const v4f*)(srow + 4);
            const float* wr = W + (((size_t)(k + 1) * CCH) + 2 * rp) * CCH + cb;
            w0 = *(const v4f*)(wr);
            w1 = *(const v4f*)(wr + CCH);
        }

        __syncthreads();   // single barrier per k (double-buffered LDS)

        // ---- 16x16x64 accumulation: 16 chained f32 WMMAs (K chunks of 4) ----
#pragma unroll
        for (int kk = 0; kk < 16; ++kk) {
            const int kc = kk * 4 + 2 * lhalf;
            // A 16x4: lane M=lm, K={kc,kc+1} (lanes 16-31 at +2)
            v2f afrag = *(const v2f*)&sa[buf][(mi + lm) * LDA + kc];
            // B 4x16: lane N=lm, rows {kc,kc+1} pre-paired -> single b64
            v2f bfrag = sb[buf][(kk * 2 + lhalf) * SBST + ni + lm];
            acc = __builtin_amdgcn_wmma_f32_16x16x4_f32(
                false, afrag, false, bfrag, (short)0, acc, false, false);
        }
    }

    // ---- epilogue: BN (scale/shift), optional residual, ReLU ----
    const int row0 = base + mi + 8 * lhalf;    // C/D layout: VGPR j -> M = j (+8 hi lanes)
#pragma unroll
    for (int j = 0; j < 8; ++j) {
        const int node = row0 + j;
        float h = acc[j] * sc + tc;
        if (resid_flag) h += resid[(size_t)node * CCH + c];
        h = fmaxf(h, 0.0f);
        dst[(size_t)node * CCH + c] = h;
    }
}

extern "C" void kernel_launch(void* const* d_in, const int* in_sizes, int n_in,
                              void* d_out, int out_size, void* d_ws, size_t ws_size,
                              hipStream_t stream) {
    (void)in_sizes; (void)n_in; (void)out_size; (void)ws_size;
    const float* x   = (const float*)d_in[0];
    const int*   nbr = (const int*)  d_in[1];
    const float* W1  = (const float*)d_in[2];
    const float* W2  = (const float*)d_in[3];
    const float* g1  = (const float*)d_in[4];
    const float* b1  = (const float*)d_in[5];
    const float* m1  = (const float*)d_in[6];
    const float* v1  = (const float*)d_in[7];
    const float* g2  = (const float*)d_in[8];
    const float* b2  = (const float*)d_in[9];
    const float* m2  = (const float*)d_in[10];
    const float* v2  = (const float*)d_in[11];

    float* h1  = (float*)d_ws;          // [N, 64] intermediate (51.2 MB)
    float* out = (float*)d_out;

    dim3 grid(NNODES / MT);             // 3125 exact
    dim3 block(THREADS);

    // stage 1: h1 = relu(bn1(conv(x, W1)))
    sparse_conv_bn_kernel<<<grid, block, 0, stream>>>(
        x, nbr, W1, g1, b1, m1, v1, nullptr, h1, /*resid=*/0);
    // stage 2: out = relu(bn2(conv(h1, W2)) + x)
    sparse_conv_bn_kernel<<<grid, block, 0, stream>>>(
        h1, nbr, W2, g2, b2, m2, v2, x, out, /*resid=*/1);
}